// LightGCN_22909355557570
// MI455X (gfx1250) — compile-verified
//
#include <hip/hip_runtime.h>
#include <hip/hip_bf16.h>
#include <stdint.h>

// ---------------- problem constants (match reference) ----------------
#define N_USERS 100000
#define N_ITEMS 50000
#define NNODES  150000          // N_USERS + N_ITEMS
#define DIM     64
#define NEDGES  6000000
#define BATCH   8192
#define NLAYERS 3
#define ETILE   256             // edges staged per LDS tile

typedef __attribute__((ext_vector_type(2))) float v2f;
typedef __attribute__((ext_vector_type(8))) float v8f;

// ---------------- CDNA5 inline-asm helpers ----------------
// Async copy of 4 bytes global -> LDS (per-lane), tracked by ASYNCcnt.
__device__ __forceinline__ void async_ld_b32(unsigned lds_off, const void* gp) {
    asm volatile("global_load_async_to_lds_b32 %0, %1, off"
                 :: "v"(lds_off), "v"((unsigned long long)(uintptr_t)gp)
                 : "memory");
}
__device__ __forceinline__ void wait_async0() {
    asm volatile("s_wait_asynccnt 0" ::: "memory");
}
// Native f32 atomic add, no return (STOREcnt), device scope so all WGPs agree.
__device__ __forceinline__ void atomic_add_f32(float* p, float v) {
    asm volatile("global_atomic_add_f32 %0, %1, off scope:SCOPE_DEV"
                 :: "v"((unsigned long long)(uintptr_t)p), "v"(v)
                 : "memory");
}

// ---------------- init: acc = emb = concat(user_emb, item_emb) ----------------
__global__ __launch_bounds__(256) void lightgcn_init(const float4* __restrict__ ue,
                                                     const float4* __restrict__ ie,
                                                     float4* __restrict__ acc,
                                                     float4* __restrict__ emb) {
    const int userQ = N_USERS * DIM / 4;   // 1,600,000
    const int totQ  = NNODES  * DIM / 4;   // 2,400,000
    for (int i = blockIdx.x * 256 + threadIdx.x; i < totQ; i += gridDim.x * 256) {
        float4 v = (i < userQ) ? ue[i] : ie[i - userQ];
        acc[i] = v;
        emb[i] = v;
    }
}

__global__ __launch_bounds__(256) void lightgcn_zero(float4* __restrict__ p) {
    const int totQ = NNODES * DIM / 4;
    float4 z; z.x = z.y = z.z = z.w = 0.0f;
    for (int i = blockIdx.x * 256 + threadIdx.x; i < totQ; i += gridDim.x * 256)
        p[i] = z;
}

__global__ __launch_bounds__(256) void lightgcn_axpy(float4* __restrict__ acc,
                                                     const float4* __restrict__ src) {
    const int totQ = NNODES * DIM / 4;
    for (int i = blockIdx.x * 256 + threadIdx.x; i < totQ; i += gridDim.x * 256) {
        float4 a = acc[i], s = src[i];
        a.x += s.x; a.y += s.y; a.z += s.z; a.w += s.w;
        acc[i] = a;
    }
}

// ---------------- SpMM scatter: eout[row] += val * ein[col] ----------------
// Edge triples double-buffered in LDS via async global->LDS loads.
// 16 lanes (4 floats each) cover one D=64 row; 256 threads = 16 edge groups.
__global__ __launch_bounds__(256) void lightgcn_spmm(const int*   __restrict__ rows,
                                                     const int*   __restrict__ cols,
                                                     const float* __restrict__ vals,
                                                     const float* __restrict__ ein,
                                                     float*       __restrict__ eout) {
    __shared__ int   s_row[2][ETILE];
    __shared__ int   s_col[2][ETILE];
    __shared__ float s_val[2][ETILE];

    const int tid      = threadIdx.x;
    const int numTiles = (NEDGES + ETILE - 1) / ETILE;   // 23438
    int t = blockIdx.x;
    if (t >= numTiles) return;

    // prefetch first tile into buffer 0
    {
        int gi = t * ETILE + tid; if (gi >= NEDGES) gi = NEDGES - 1;
        async_ld_b32((unsigned)(uintptr_t)&s_row[0][tid], rows + gi);
        async_ld_b32((unsigned)(uintptr_t)&s_col[0][tid], cols + gi);
        async_ld_b32((unsigned)(uintptr_t)&s_val[0][tid], vals + gi);
    }

    int pb = 0;
    const int g   = tid >> 4;   // edge group 0..15
    const int sub = tid & 15;   // float4 slot within row

    while (t < numTiles) {
        wait_async0();          // my wave's async loads for buffer pb complete
        __syncthreads();        // all waves' loads complete; prior reads of pb^1 done

        const int tn = t + gridDim.x;
        if (tn < numTiles) {    // prefetch next tile into the other buffer
            int gi = tn * ETILE + tid; if (gi >= NEDGES) gi = NEDGES - 1;
            const int nb = pb ^ 1;
            async_ld_b32((unsigned)(uintptr_t)&s_row[nb][tid], rows + gi);
            async_ld_b32((unsigned)(uintptr_t)&s_col[nb][tid], cols + gi);
            async_ld_b32((unsigned)(uintptr_t)&s_val[nb][tid], vals + gi);
        }

        const int base = t * ETILE;
        #pragma unroll 4
        for (int k = 0; k < 16; ++k) {
            const int e = g * 16 + k;
            if (base + e < NEDGES) {
                const int   r = s_row[pb][e];
                const int   c = s_col[pb][e];
                const float v = s_val[pb][e];
                const float4 x = *(const float4*)(ein + (size_t)c * DIM + sub * 4);
                float* dst = eout + (size_t)r * DIM + sub * 4;
                atomic_add_f32(dst + 0, v * x.x);
                atomic_add_f32(dst + 1, v * x.y);
                atomic_add_f32(dst + 2, v * x.z);
                atomic_add_f32(dst + 3, v * x.w);
            }
        }
        pb ^= 1;
        t = tn;
    }
}

// ---------------- acc *= 0.25 via V_WMMA_F32_16X16X4_F32 ----------------
// Tile T (16x16) scaled exactly: D = sum_i T[:,4i:4i+4] x B_i, where
// B_i[k,n] = 0.25 iff n == 4i+k. Each output element gets exactly one
// product (0.25 is a power of two) => bit-exact acc/4.
__global__ __launch_bounds__(256) void lightgcn_scale_wmma(float* __restrict__ acc) {
    const int wave = threadIdx.x >> 5;
    const int lane = threadIdx.x & 31;
    const int tile = blockIdx.x * 8 + wave;       // one 16x16 tile per wave
    const int colTiles = DIM / 16;                // 4
    const int total    = (NNODES / 16) * colTiles;// 37500
    if (tile >= total) return;                    // wave-uniform: EXEC stays all-1s

    const int rt = tile / colTiles;
    const int ct = tile % colTiles;
    const int l  = lane & 15;
    const int hi = lane >> 4;                     // 0: K/M rows 0..1 ; 1: 2..3
    float* T = acc + (size_t)(rt * 16) * DIM + ct * 16;

    v8f c = {0.f, 0.f, 0.f, 0.f, 0.f, 0.f, 0.f, 0.f};
    #pragma unroll
    for (int i = 0; i < 4; ++i) {
        const int h = hi ? 2 : 0;
        v2f a, b;
        // A layout (16x4 f32): lane l holds M=l; VGPR0 = K = h, VGPR1 = K = h+1
        a.x = T[(size_t)l * DIM + 4 * i + h + 0];
        a.y = T[(size_t)l * DIM + 4 * i + h + 1];
        // B layout (4x16 f32): lane l is column N=l; VGPR0 = K = h, VGPR1 = K = h+1
        b.x = (l == 4 * i + h + 0) ? 0.25f : 0.0f;
        b.y = (l == 4 * i + h + 1) ? 0.25f : 0.0f;
        c = __builtin_amdgcn_wmma_f32_16x16x4_f32(false, a, false, b,
                                                  (short)0, c, false, false);
    }
    // C/D layout: VGPR r -> row (r + hi*8), column l
    #pragma unroll
    for (int r = 0; r < 8; ++r) {
        const int m = r + hi * 8;
        T[(size_t)m * DIM + l] = c[r];
    }
}

// ---------------- fused 6-way gather into d_out ----------------
__global__ __launch_bounds__(256) void lightgcn_gather(const int* __restrict__ users,
                                                       const int* __restrict__ pos,
                                                       const int* __restrict__ neg,
                                                       const float* __restrict__ light,
                                                       const float* __restrict__ ue,
                                                       const float* __restrict__ ie,
                                                       float* __restrict__ out) {
    const int gt    = blockIdx.x * 256 + threadIdx.x;  // 6*8192*16 threads
    const int rowid = gt >> 4;
    const int sub   = gt & 15;
    if (rowid >= 6 * BATCH) return;
    const int sel = rowid / BATCH;
    const int b   = rowid % BATCH;
    const float* src;
    switch (sel) {
        case 0:  src = light + (size_t)users[b] * DIM;             break;
        case 1:  src = light + (size_t)(N_USERS + pos[b]) * DIM;   break;
        case 2:  src = light + (size_t)(N_USERS + neg[b]) * DIM;   break;
        case 3:  src = ue    + (size_t)users[b] * DIM;             break;
        case 4:  src = ie    + (size_t)pos[b] * DIM;               break;
        default: src = ie    + (size_t)neg[b] * DIM;               break;
    }
    const float4 v = *(const float4*)(src + sub * 4);
    *(float4*)(out + (size_t)rowid * DIM + sub * 4) = v;
}

// ---------------- launcher ----------------
extern "C" void kernel_launch(void* const* d_in, const int* in_sizes, int n_in,
                              void* d_out, int out_size, void* d_ws, size_t ws_size,
                              hipStream_t stream) {
    (void)in_sizes; (void)n_in; (void)out_size; (void)ws_size;
    const int*   users = (const int*)  d_in[0];
    const int*   pos   = (const int*)  d_in[1];
    const int*   neg   = (const int*)  d_in[2];
    const int*   grow  = (const int*)  d_in[3];
    const int*   gcol  = (const int*)  d_in[4];
    const float* gval  = (const float*)d_in[5];
    const float* ue    = (const float*)d_in[6];
    const float* ie    = (const float*)d_in[7];
    float* out = (float*)d_out;

    const size_t bufElems = (size_t)NNODES * DIM;     // 9.6M floats = 38.4 MB
    float* acc  = (float*)d_ws;
    float* embA = acc  + bufElems;
    float* embB = embA + bufElems;

    const int Q = (int)(bufElems / 4);                // float4 count
    const dim3 blk(256);
    const dim3 gridQ((Q + 255) / 256);

    lightgcn_init<<<gridQ, blk, 0, stream>>>((const float4*)ue, (const float4*)ie,
                                             (float4*)acc, (float4*)embA);

    float* src = embA;
    float* dst = embB;
    for (int layer = 0; layer < NLAYERS; ++layer) {
        lightgcn_zero<<<gridQ, blk, 0, stream>>>((float4*)dst);
        lightgcn_spmm<<<dim3(4096), blk, 0, stream>>>(grow, gcol, gval, src, dst);
        lightgcn_axpy<<<gridQ, blk, 0, stream>>>((float4*)acc, (const float4*)dst);
        float* tmp = src; src = dst; dst = tmp;
    }

    const int wmmaTiles = (NNODES / 16) * (DIM / 16); // 37500
    lightgcn_scale_wmma<<<dim3((wmmaTiles + 7) / 8), blk, 0, stream>>>(acc);

    lightgcn_gather<<<dim3(6 * BATCH * 16 / 256), blk, 0, stream>>>(
        users, pos, neg, acc, ue, ie, out);
}